// GCN_13056700580576
// MI455X (gfx1250) — compile-verified
//
#include <hip/hip_runtime.h>

typedef __attribute__((ext_vector_type(2))) float v2f;
typedef __attribute__((ext_vector_type(8))) float v8f;

#define KDIM 512
#define F1 32
#define F2 16

// ---------------- init: deg=1 (self loop), agg1=0, out=0 ----------------
__global__ void init_kernel(float* __restrict__ deg, float* __restrict__ agg1,
                            float* __restrict__ out, int n) {
  int i = blockIdx.x * blockDim.x + threadIdx.x;
  const int t1 = n, t2 = n + n * F1, t3 = n + n * F1 + n * F2;
  if (i < t1)       deg[i] = 1.0f;
  else if (i < t2)  agg1[i - t1] = 0.0f;
  else if (i < t3)  out[i - t2] = 0.0f;
}

// ---------------- degree: deg[dst[e]] += 1 ----------------
__global__ void degree_kernel(const int* __restrict__ dst, float* __restrict__ deg, int E) {
  int i = blockIdx.x * blockDim.x + threadIdx.x;
  if (i < E) atomicAdd(&deg[dst[i]], 1.0f);
}

// ---------------- dis = rsqrt(deg), in place ----------------
__global__ void rsqrt_kernel(float* __restrict__ deg, int n) {
  int i = blockIdx.x * blockDim.x + threadIdx.x;
  if (i < n) deg[i] = rsqrtf(deg[i]);
}

// ---------------- GEMM1: h[n,32] = x[n,512] @ W1[512,32]  (f32 WMMA) ----
__global__ __launch_bounds__(256) void gemm1_kernel(const float* __restrict__ x,
                                                    const float* __restrict__ W1,
                                                    float* __restrict__ h, int n) {
  const int lane  = threadIdx.x & 31;
  const int wave  = blockIdx.x * (blockDim.x >> 5) + (threadIdx.x >> 5);
  const int ntile = n >> 4;                      // n is a multiple of 16
  if (wave >= ntile) return;                     // wave-uniform: EXEC stays all-1s
  const int m0 = wave << 4;
  const int r  = lane & 15;
  const int kh = (lane >> 4) << 1;               // 0 or 2 (K sub-offset)

  const float* arow = x + (size_t)(m0 + r) * KDIM;

  v8f acc0 = {0.f,0.f,0.f,0.f,0.f,0.f,0.f,0.f};
  v8f acc1 = {0.f,0.f,0.f,0.f,0.f,0.f,0.f,0.f};

  for (int k = 0; k < KDIM; k += 4) {
    // A fragment: x[m0+r][k+kh .. k+kh+1], streamed (non-temporal: keep L2 for h/agg)
    v2f a = __builtin_nontemporal_load((const v2f*)(arow + k + kh));
    // B fragments: W1 rows (k+kh), (k+kh+1), cols r and r+16 (cached, tiny)
    const float* w0 = W1 + (size_t)(k + kh) * F1;
    const float* w1 = w0 + F1;
    v2f bA; bA.x = w0[r];      bA.y = w1[r];
    v2f bB; bB.x = w0[r + 16]; bB.y = w1[r + 16];
    acc0 = __builtin_amdgcn_wmma_f32_16x16x4_f32(false, a, false, bA, (short)0, acc0, false, false);
    acc1 = __builtin_amdgcn_wmma_f32_16x16x4_f32(false, a, false, bB, (short)0, acc1, false, false);
  }

  // C/D layout: VGPR v, lanes 0-15 -> M=v, lanes 16-31 -> M=8+v, N=lane&15
  const int mbase = m0 + ((lane >> 4) << 3);
  float* hp = h + (size_t)mbase * F1 + r;
#pragma unroll
  for (int v = 0; v < 8; ++v) {
    hp[(size_t)v * F1]      = acc0[v];
    hp[(size_t)v * F1 + 16] = acc1[v];
  }
}

// ---- GEMM2 (fused layer-1 epilogue):
//      a[row,k] = relu(agg1[row,k] + h[row,k]*dis[row]^2 + b1[k]);  h2 = a @ W2
__global__ __launch_bounds__(256) void gemm2_kernel(const float* __restrict__ agg1,
                                                    const float* __restrict__ h,
                                                    const float* __restrict__ dis,
                                                    const float* __restrict__ b1,
                                                    const float* __restrict__ W2,
                                                    float* __restrict__ h2, int n) {
  const int lane  = threadIdx.x & 31;
  const int wave  = blockIdx.x * (blockDim.x >> 5) + (threadIdx.x >> 5);
  const int ntile = n >> 4;
  if (wave >= ntile) return;
  const int m0 = wave << 4;
  const int r  = lane & 15;
  const int kh = (lane >> 4) << 1;

  const int row  = m0 + r;
  const float dv = dis[row];
  const float d2 = dv * dv;
  const float* ag = agg1 + (size_t)row * F1;
  const float* hr = h    + (size_t)row * F1;

  v8f acc = {0.f,0.f,0.f,0.f,0.f,0.f,0.f,0.f};

  for (int k = 0; k < F1; k += 4) {
    const int k0 = k + kh;
    v2f a;
    a.x = fmaxf(ag[k0]     + hr[k0]     * d2 + b1[k0],     0.0f);
    a.y = fmaxf(ag[k0 + 1] + hr[k0 + 1] * d2 + b1[k0 + 1], 0.0f);
    v2f b;
    b.x = W2[(size_t)k0 * F2 + r];
    b.y = W2[(size_t)(k0 + 1) * F2 + r];
    acc = __builtin_amdgcn_wmma_f32_16x16x4_f32(false, a, false, b, (short)0, acc, false, false);
  }

  const int mbase = m0 + ((lane >> 4) << 3);
  float* op = h2 + (size_t)mbase * F2 + r;
#pragma unroll
  for (int v = 0; v < 8; ++v) op[(size_t)v * F2] = acc[v];
}

// ---- edge aggregation: out[dst] += feat[src] * dis[src]*dis[dst]
// Wave32 processes 32 edges; lane = feature. Gathers and atomic bursts hit one
// cacheline per edge (everything L2-resident on MI455X's 192 MB L2).
template <int F>
__global__ __launch_bounds__(256) void agg_kernel(const float* __restrict__ feat,
                                                  const int* __restrict__ src,
                                                  const int* __restrict__ dst,
                                                  const float* __restrict__ dis,
                                                  float* __restrict__ out, int E) {
  const int lane = threadIdx.x & 31;
  const int wave = blockIdx.x * (blockDim.x >> 5) + (threadIdx.x >> 5);
  const int e0   = wave * 32;
  if (e0 >= E) return;

  const int myE = e0 + lane;
  int   s_l = -1, d_l = 0;
  float nrm_l = 0.0f;
  if (myE < E) {
    s_l = src[myE];
    d_l = dst[myE];
    nrm_l = dis[s_l] * dis[d_l];
  }

  constexpr int EPS = 32 / F;       // edges handled per step (1 for F=32, 2 for F=16)
  const int f   = lane & (F - 1);
  const int sub = lane / F;

  for (int j = 0; j < 32; j += EPS) {
    const int   s  = __shfl(s_l,   j + sub);
    const int   d  = __shfl(d_l,   j + sub);
    const float nm = __shfl(nrm_l, j + sub);
    if (s >= 0)
      atomicAdd(&out[(size_t)d * F + f], feat[(size_t)s * F + f] * nm);
  }
}

// ---- final epilogue: out = agg2 + h2*dis^2 + b2 (in place on d_out) ----
__global__ void epi2_kernel(float* __restrict__ out, const float* __restrict__ h2,
                            const float* __restrict__ dis, const float* __restrict__ b2,
                            int n) {
  int i = blockIdx.x * blockDim.x + threadIdx.x;
  if (i < n * F2) {
    const int row = i / F2, f = i - row * F2;
    const float dv = dis[row];
    out[i] = out[i] + h2[i] * dv * dv + b2[f];
  }
}

extern "C" void kernel_launch(void* const* d_in, const int* in_sizes, int n_in,
                              void* d_out, int out_size, void* d_ws, size_t ws_size,
                              hipStream_t stream) {
  const float* x  = (const float*)d_in[0];
  const int*   ei = (const int*)d_in[1];
  const float* W1 = (const float*)d_in[2];
  const float* b1 = (const float*)d_in[3];
  const float* W2 = (const float*)d_in[4];
  const float* b2 = (const float*)d_in[5];

  const int n = in_sizes[0] / KDIM;   // 100000
  const int E = in_sizes[1] / 2;      // 3200000
  const int* src = ei;
  const int* dst = ei + E;

  float* ws   = (float*)d_ws;
  float* dis  = ws;                         // n       (deg -> rsqrt in place)
  float* h    = dis  + n;                   // n*32
  float* agg1 = h    + (size_t)n * F1;      // n*32
  float* h2   = agg1 + (size_t)n * F1;      // n*16
  float* out  = (float*)d_out;              // n*16, used as layer-2 accumulator

  // 1) init deg/agg1/out
  {
    const int total = n + n * F1 + n * F2;
    init_kernel<<<(total + 255) / 256, 256, 0, stream>>>(dis, agg1, out, n);
  }
  // 2) in-degree
  degree_kernel<<<(E + 255) / 256, 256, 0, stream>>>(dst, dis, E);
  // 3) dis = deg^-1/2
  rsqrt_kernel<<<(n + 255) / 256, 256, 0, stream>>>(dis, n);
  // 4) h = x @ W1 (WMMA f32)
  {
    const int ntile = n >> 4;                    // 6250 wave tiles
    gemm1_kernel<<<(ntile + 7) / 8, 256, 0, stream>>>(x, W1, h, n);
  }
  // 5) agg1[d] += h[s]*norm
  {
    const int waves = (E + 31) / 32;
    agg_kernel<F1><<<(waves + 7) / 8, 256, 0, stream>>>(h, src, dst, dis, agg1, E);
  }
  // 6) h2 = relu(agg1 + h*dis^2 + b1) @ W2 (fused epilogue + WMMA f32)
  {
    const int ntile = n >> 4;
    gemm2_kernel<<<(ntile + 7) / 8, 256, 0, stream>>>(agg1, h, dis, b1, W2, h2, n);
  }
  // 7) out[d] += h2[s]*norm
  {
    const int waves = (E + 31) / 32;
    agg_kernel<F2><<<(waves + 7) / 8, 256, 0, stream>>>(h2, src, dst, dis, out, E);
  }
  // 8) out += h2*dis^2 + b2
  epi2_kernel<<<(n * F2 + 255) / 256, 256, 0, stream>>>(out, h2, dis, b2, n);
}